// DecoderTreeLSTMCell_52183852646692
// MI455X (gfx1250) — compile-verified
//
#include <hip/hip_runtime.h>
#include <math.h>

#define N_POS 10
#define H 128
#define WFULL 512  // fused output width: [ u(128) | i(128) | o(128) | uu(128) ]
#define LDA 132    // LDS stride for A tile (floats) — bank-conflict-free fragment reads
#define LDO 516    // LDS stride for output tile (floats) — de-conflicts lanes 16-31 stores

typedef __attribute__((ext_vector_type(2))) float v2f;
typedef __attribute__((ext_vector_type(8))) float v8f;

__global__ void zero_counts_kernel(int* __restrict__ cnt) {
    int t = threadIdx.x;
    if (t < N_POS) cnt[t] = 0;
}

__global__ void bin_rows_kernel(const int* __restrict__ pos,
                                int* __restrict__ cnt,
                                int* __restrict__ idx,
                                int N) {
    int n = blockIdx.x * blockDim.x + threadIdx.x;
    if (n >= N) return;
    int p = pos[n];
    int slot = atomicAdd(&cnt[p], 1);
    idx[p * N + slot] = n;
}

// Build W_fused[p][k][jj] (k-major, jj contiguous) so every B-fragment load in
// the GEMM is a coalesced 64B segment per half-wave:
//   jj <  H : W_f[(p*H + k)*H + jj]                 (u gate)
//   jj >= H : W_iou[(jj-H)*(N_POS*H) + p*H + k]     (transposed i|o|uu)
__global__ void fuse_weights_kernel(const float* __restrict__ W_f,
                                    const float* __restrict__ W_iou,
                                    float* __restrict__ W_fused) {
    int t = blockIdx.x * blockDim.x + threadIdx.x;
    if (t >= N_POS * H * WFULL) return;
    const int jj = t & (WFULL - 1);
    const int k  = (t >> 9) & (H - 1);
    const int p  = t >> 16;
    float v;
    if (jj < H) v = W_f[((size_t)(p * H + k)) * H + jj];
    else        v = W_iou[(size_t)(jj - H) * (N_POS * H) + p * H + k];
    W_fused[t] = v;
}

__device__ __forceinline__ float sigmoidf_(float x) {
    return 1.0f / (1.0f + expf(-x));
}

// One workgroup = one pos p + one 16-row tile of that pos-bin.
__global__ __launch_bounds__(256)
void treelstm_gemm_kernel(const float* __restrict__ child_h,
                          const float* __restrict__ child_c,
                          const float* __restrict__ e1,
                          const float* __restrict__ e2,
                          const float* __restrict__ h_prev,
                          const int*   __restrict__ depth,
                          const int*   __restrict__ mask,
                          const float* __restrict__ W_fused, // [N_POS][H][WFULL]
                          const float* __restrict__ b_f,     // [N_POS*H]
                          const float* __restrict__ b_iou,   // [3H]
                          const int*   __restrict__ cnt,
                          const int*   __restrict__ idx,
                          float* __restrict__ out,           // [2, N, H] : h then c
                          int N) {
    const int p = blockIdx.x;
    const int count = cnt[p];
    const int tile0 = blockIdx.y * 16;
    if (tile0 >= count) return;

    __shared__ float lds_a[16 * LDA];   // h_cat tile, 16 x 128
    __shared__ float lds_o[16 * LDO];   // fused output tile, 16 x 512

    const int tid = threadIdx.x;

    // ---- Stage A tile (gathered, depth-injected h_cat) into LDS ----
    {
        const int r  = tid >> 4;          // 0..15 row within tile
        const int c0 = (tid & 15) * 8;    // 8 contiguous cols per thread
        const int g  = tile0 + r;
        if (g < count) {
            const int row = idx[p * N + g];
            const int d   = depth[row];
            const float s1 = (d == 1) ? 1.0f : 0.0f;
            const float s2 = (d == 2) ? 1.0f : 0.0f;
            const float* __restrict__ ch = child_h + (size_t)row * H;
            const float* __restrict__ x1 = e1 + (size_t)row * H;
            const float* __restrict__ x2 = e2 + (size_t)row * H;
            #pragma unroll
            for (int q = 0; q < 8; ++q) {
                int cc = c0 + q;
                lds_a[r * LDA + cc] = ch[cc] + s1 * x1[cc] + s2 * x2[cc];
            }
        } else {
            #pragma unroll
            for (int q = 0; q < 8; ++q)
                lds_a[r * LDA + c0 + q] = 0.0f;
        }
    }
    __syncthreads();

    // ---- GEMM: 8 waves; each wave owns 4 adjacent 16-wide column tiles.
    // Single K-loop with 4 interleaved accumulator chains: A fragment shared
    // across 4 WMMAs, dependency distance 4 between chained WMMAs.
    const int wave = tid >> 5;
    const int lane = tid & 31;
    const int mrow = lane & 15;             // M for A frag / N for B,C frags
    const int kk   = (lane < 16) ? 0 : 2;   // K sub-offset per documented layout

    v8f acc[4];
    #pragma unroll
    for (int i = 0; i < 4; ++i)
        acc[i] = (v8f){0.f, 0.f, 0.f, 0.f, 0.f, 0.f, 0.f, 0.f};

    // B base: column block jj0 = wave*64, this lane's column = jj0 + i*16 + mrow
    const float* __restrict__ bb =
        W_fused + (size_t)p * H * WFULL + wave * 64 + mrow;

    #pragma unroll
    for (int k0 = 0; k0 < H; k0 += 4) {
        v2f a;
        a.x = lds_a[mrow * LDA + k0 + kk];
        a.y = lds_a[mrow * LDA + k0 + kk + 1];
        #pragma unroll
        for (int i = 0; i < 4; ++i) {
            v2f b;
            b.x = bb[(size_t)(k0 + kk) * WFULL + i * 16];
            b.y = bb[(size_t)(k0 + kk + 1) * WFULL + i * 16];
            acc[i] = __builtin_amdgcn_wmma_f32_16x16x4_f32(
                         false, a, false, b, (short)0, acc[i], false, false);
        }
    }

    // C/D layout: VGPR v -> row (v + 8*(lane>=16)), col = lane&15
    #pragma unroll
    for (int i = 0; i < 4; ++i) {
        const int ct = wave * 4 + i;
        #pragma unroll
        for (int v = 0; v < 8; ++v) {
            const int m = v + ((lane < 16) ? 0 : 8);
            lds_o[m * LDO + ct * 16 + mrow] = acc[i][v];
        }
    }
    __syncthreads();

    // ---- Fused LSTM epilogue + mask blend ----
    for (int e = tid; e < 16 * H; e += 256) {
        const int r = e >> 7;        // tile row
        const int j = e & (H - 1);   // feature
        const int g = tile0 + r;
        if (g >= count) continue;
        const int row = idx[p * N + g];

        const float u_v  = lds_o[r * LDO +         j] + b_f[p * H + j];
        const float i_v  = lds_o[r * LDO + 1 * H + j] + b_iou[j];
        const float o_v  = lds_o[r * LDO + 2 * H + j] + b_iou[H + j];
        const float uu_v = lds_o[r * LDO + 3 * H + j] + b_iou[2 * H + j];

        const float f     = sigmoidf_(u_v);
        const float c_red = f * child_c[(size_t)row * H + j];
        const float c_new = sigmoidf_(i_v) * tanhf(uu_v) + c_red;
        const float h_new = sigmoidf_(o_v) * tanhf(c_new);

        const float m = (float)mask[row];
        out[(size_t)row * H + j]                 = h_prev[(size_t)row * H + j] * m + (1.0f - m) * h_new;
        out[(size_t)N * H + (size_t)row * H + j] = c_red * m + (1.0f - m) * c_new;
    }
}

extern "C" void kernel_launch(void* const* d_in, const int* in_sizes, int n_in,
                              void* d_out, int out_size, void* d_ws, size_t ws_size,
                              hipStream_t stream) {
    const float* child_h = (const float*)d_in[0];
    const float* child_c = (const float*)d_in[1];
    const float* e1      = (const float*)d_in[2];
    const float* e2      = (const float*)d_in[3];
    const float* h_prev  = (const float*)d_in[4];
    const int*   pos     = (const int*)d_in[5];
    const int*   depth   = (const int*)d_in[6];
    const int*   mask    = (const int*)d_in[7];
    const float* W_f     = (const float*)d_in[8];
    const float* b_f     = (const float*)d_in[9];
    const float* W_iou   = (const float*)d_in[10];
    const float* b_iou   = (const float*)d_in[11];
    float* out = (float*)d_out;

    const int N = in_sizes[5];   // rows (pos has one entry per row)

    // Workspace layout: [cnt: 64 ints][idx: N_POS*N ints][W_fused: N_POS*H*WFULL floats]
    int*   wsi     = (int*)d_ws;
    int*   cnt     = wsi;
    int*   idx     = wsi + 64;
    float* W_fused = (float*)(wsi + 64 + N_POS * N);

    zero_counts_kernel<<<1, 32, 0, stream>>>(cnt);
    bin_rows_kernel<<<(N + 255) / 256, 256, 0, stream>>>(pos, cnt, idx, N);

    const int wtot = N_POS * H * WFULL;
    fuse_weights_kernel<<<(wtot + 255) / 256, 256, 0, stream>>>(W_f, W_iou, W_fused);

    dim3 grid(N_POS, (N + 15) / 16);
    treelstm_gemm_kernel<<<grid, 256, 0, stream>>>(
        child_h, child_c, e1, e2, h_prev, depth, mask,
        W_fused, b_f, b_iou, cnt, idx, out, N);
}